// SGAPSMAELoss_52398601011855
// MI455X (gfx1250) — compile-verified
//
#include <hip/hip_runtime.h>

// ---------------------------------------------------------------------------
// SGAPS-MAE loss for MI455X (gfx1250, wave32).
// Memory-bound (~117MB -> ~5us floor @ 23.3TB/s). SSIM 11x11 box filters are
// banded-ones GEMMs on V_WMMA_F32_16X16X4_F32 (exact in f32). Patch fill uses
// the Tensor Data Mover (TENSOR_LOAD_TO_LDS, TENSORcnt) for interior tiles and
// GLOBAL_LOAD_ASYNC_TO_LDS_B32 (ASYNCcnt) for edge tiles. Perceptual |p-t| is
// fused into the same streaming pass; halo re-reads hit the 192MB L2.
// ---------------------------------------------------------------------------

typedef float v2f __attribute__((ext_vector_type(2)));
typedef float v8f __attribute__((ext_vector_type(8)));
typedef unsigned int u32x4 __attribute__((ext_vector_type(4)));
typedef int i32x4 __attribute__((ext_vector_type(4)));
typedef int i32x8 __attribute__((ext_vector_type(8)));

#define AS1 __attribute__((address_space(1)))
#define AS3 __attribute__((address_space(3)))
typedef AS1 int g_i32;   // global-memory int (async-LDS builtin operand type)
typedef AS3 int l_i32;   // LDS int

#define HW 512
#define PLANE (HW * HW)              // 262144
#define NPLANES 48                   // B*C = 16*3
#define TOTAL_TILES (NPLANES * 1024) // 49152 tiles of 16x16
#define WAVES_PER_BLOCK 4
#define BLOCK_THREADS (WAVES_PER_BLOCK * 32)
#define PR 32                        // patch rows allocated (26 valid)
#define PC 28                        // patch cols allocated == TDM tile_dim0
#define TOTAL_ELEMS 12582912.0       // 16*3*512*512
#define SSIM_C1 0.0001f
#define SSIM_C2 0.0009f

#if defined(__has_builtin)
#if __has_builtin(__builtin_amdgcn_global_load_async_to_lds_b32)
#define HAVE_ASYNC_LDS 1
#endif
#if __has_builtin(__builtin_amdgcn_tensor_load_to_lds)
#define HAVE_TDM 1
#endif
#endif

__device__ __forceinline__ void wait_asynccnt0() {
#if defined(__has_builtin)
#if __has_builtin(__builtin_amdgcn_s_wait_asynccnt)
  __builtin_amdgcn_s_wait_asynccnt(0);
  return;
#else
  asm volatile("s_wait_asynccnt 0" ::: "memory");
  return;
#endif
#endif
}

#if defined(HAVE_TDM)
// One TDM descriptor moves a 28x28 f32 tile (rows contiguous in LDS, matching
// the PC=28 patch stride). tensor_dim = elements remaining from the tile start
// so right/bottom OOB reads return zero (TDM OOB rule). Requires r0,c0 >= 0.
__device__ __forceinline__ void tdm_load_tile(const float* gsrc,
                                              unsigned lds_byte, int tdim0,
                                              int tdim1) {
  u32x4 g0;
  g0[0] = 1u;                                   // count=1 (valid user D#)
  g0[1] = lds_byte;                             // lds_addr [63:32]
  const unsigned long long ga = (unsigned long long)(uintptr_t)gsrc;
  g0[2] = (unsigned)(ga & 0xFFFFFFFFu);         // global_addr [95:64]
  g0[3] = (unsigned)((ga >> 32) & 0x1FFFFFFu)   // global_addr [120:96]
          | (2u << 30);                         // type=2 ("image")
  i32x8 g1;
  g1[0] = 0x00020000;                           // data_size=2 (4B); mask=0
  g1[1] = (tdim0 & 0xFFFF) << 16;               // tensor_dim0[15:0] @ bit48
  g1[2] = ((tdim0 >> 16) & 0xFFFF) | ((tdim1 & 0xFFFF) << 16);
  g1[3] = ((tdim1 >> 16) & 0xFFFF) | (PC << 16); // tile_dim0 = 28
  g1[4] = PC;                                   // tile_dim1 = 28, tile_dim2=0
  g1[5] = HW;                                   // tensor_dim0_stride = 512
  g1[6] = 0;                                    // stride hi / dim1_stride lo
  g1[7] = 0;
  const i32x4 z4 = {0, 0, 0, 0};                // groups 2/3 unused (2D)
  const i32x8 z8 = {0, 0, 0, 0, 0, 0, 0, 0};    // extra group (6-arg form)
  __builtin_amdgcn_tensor_load_to_lds(g0, g1, z4, z4, z8, 0);
}
#endif

__device__ __forceinline__ v8f wmma4(v2f a, v2f b, v8f c) {
  // D = A(16x4,f32) * B(4x16,f32) + C(16x16,f32)
  return __builtin_amdgcn_wmma_f32_16x16x4_f32(
      /*neg_a=*/false, a, /*neg_b=*/false, b,
      /*c_mod=*/(short)0, c, /*reuse_a=*/false, /*reuse_b=*/false);
}

__device__ __forceinline__ float mapv(float p, float t, int m) {
  if (m == 0) return p;
  if (m == 1) return t;
  if (m == 2) return p * p;
  if (m == 3) return t * t;
  return p * t;
}

// ---------------------------------------------------------------------------
// Kernel: zero the accumulators (ws[0]=sum|p-t|, ws[1]=sum ssim, ws[2]=sampled)
// ---------------------------------------------------------------------------
__global__ void init_accum(double* acc) {
  if (threadIdx.x < 3) acc[threadIdx.x] = 0.0;
}

// ---------------------------------------------------------------------------
// Kernel: SSIM (WMMA box filters) + fused perceptual |p-t| sum.
// One 16x16 output tile per wave; 4 waves per block.
// ---------------------------------------------------------------------------
__global__ __launch_bounds__(BLOCK_THREADS) void ssim_tile_kernel(
    const float* __restrict__ pred, const float* __restrict__ targ,
    double* __restrict__ acc) {
  __shared__ float Pp[WAVES_PER_BLOCK][PR][PC];
  __shared__ float Pt[WAVES_PER_BLOCK][PR][PC];
  __shared__ float Hb[WAVES_PER_BLOCK][PR][17];   // horizontal-pass result
  __shared__ float red[BLOCK_THREADS];

  const int lane = threadIdx.x & 31;
  const int wave = threadIdx.x >> 5;
  const int tile = blockIdx.x * WAVES_PER_BLOCK + wave;
  const int plane = tile >> 10;          // tile / 1024
  const int trem  = tile & 1023;
  const int ty = trem >> 5;              // tile row in plane
  const int tx = trem & 31;              // tile col in plane
  const int r0 = ty * 16 - 5;
  const int c0 = tx * 16 - 5;
  const size_t pbase = (size_t)plane * PLANE;

  // ---- patch fill (zero-padded).
#if defined(HAVE_TDM)
  if (r0 >= 0 && c0 >= 0) {
    // Interior / right / bottom tiles: one TDM descriptor per tensor.
    // Wave-uniform branch; TDM ignores EXEC and is wave-scoped.
    const size_t goff = pbase + (size_t)r0 * HW + c0;
    tdm_load_tile(pred + goff, (unsigned)(uintptr_t)&Pp[wave][0][0],
                  HW - c0, HW - r0);
    tdm_load_tile(targ + goff, (unsigned)(uintptr_t)&Pt[wave][0][0],
                  HW - c0, HW - r0);
    // Rows 28..31 stay stale: they feed only H rows that are never read.
    __builtin_amdgcn_s_wait_tensorcnt(0);
  } else
#endif
  {
#if defined(HAVE_ASYNC_LDS)
    for (int i = lane; i < PR * PC; i += 32) {
      const int r = i / PC, c = i % PC;
      const int gr = r0 + r, gc = c0 + c;
      const bool ok = (r < 26) && (c < 26) && (gr >= 0) && (gr < HW) &&
                      (gc >= 0) && (gc < HW);
      if (ok) {
        const size_t off = pbase + (size_t)gr * HW + gc;
        // memory -> LDS without a VGPR round-trip (ASYNCcnt tracked)
        __builtin_amdgcn_global_load_async_to_lds_b32(
            (g_i32*)(uintptr_t)(pred + off),
            (l_i32*)(unsigned int)(uintptr_t)&Pp[wave][r][c], 0, 0);
        __builtin_amdgcn_global_load_async_to_lds_b32(
            (g_i32*)(uintptr_t)(targ + off),
            (l_i32*)(unsigned int)(uintptr_t)&Pt[wave][r][c], 0, 0);
      } else {
        // pad positions are disjoint from async targets -> no ordering hazard
        Pp[wave][r][c] = 0.0f;
        Pt[wave][r][c] = 0.0f;
      }
    }
    wait_asynccnt0();
#else
    for (int i = lane; i < PR * PC; i += 32) {
      const int r = i / PC, c = i % PC;
      const int gr = r0 + r, gc = c0 + c;
      const bool ok = (r < 26) && (c < 26) && (gr >= 0) && (gr < HW) &&
                      (gc >= 0) && (gc < HW);
      const size_t off = pbase + (size_t)(ok ? gr : 0) * HW + (ok ? gc : 0);
      Pp[wave][r][c] = ok ? pred[off] : 0.0f;
      Pt[wave][r][c] = ok ? targ[off] : 0.0f;
    }
#endif
  }
  __syncthreads();

  // Per-lane fragment coordinates (gfx1250 wave32 WMMA layouts, ISA 7.12.2):
  //   A 16x4 f32:  M = lane%16, K(slot s) = s + 2*(lane/16)
  //   B 4x16 f32:  K(slot s) = s + 2*(lane/16), N = lane%16
  //   C/D 16x16:   row(vgpr v) = v + 8*(lane/16), col = lane%16
  const int n = lane & 15;
  const int khalf = (lane >> 4) << 1;
  const int rowoff = (lane >> 4) << 3;   // 8*(lane/16)

  // Band-matrix fragments: Sh[k][n] (as B) == Sv[n][k] (as A) == [n <= k <= n+10].
  v2f band[7];
#pragma unroll
  for (int kb = 0; kb < 7; ++kb) {
    const int k0 = kb * 4 + khalf;
    const int k1 = k0 + 1;
    band[kb].x = (k0 >= n && k0 <= n + 10) ? 1.0f : 0.0f;
    band[kb].y = (k1 >= n && k1 <= n + 10) ? 1.0f : 0.0f;
  }

  v8f O[5];

#pragma unroll
  for (int m = 0; m < 5; ++m) {
    // ---- horizontal pass: H(32x16) = Patch(32x28) * Sh(28x16)
    v8f h0 = {0.f, 0.f, 0.f, 0.f, 0.f, 0.f, 0.f, 0.f};
    v8f h1 = h0;
#pragma unroll
    for (int kb = 0; kb < 7; ++kb) {
      const int k0 = kb * 4 + khalf;
      const int k1 = k0 + 1;
      v2f a0, a1;
      a0.x = mapv(Pp[wave][n][k0],      Pt[wave][n][k0],      m);
      a0.y = mapv(Pp[wave][n][k1],      Pt[wave][n][k1],      m);
      a1.x = mapv(Pp[wave][16 + n][k0], Pt[wave][16 + n][k0], m);
      a1.y = mapv(Pp[wave][16 + n][k1], Pt[wave][16 + n][k1], m);
      h0 = wmma4(a0, band[kb], h0);
      h1 = wmma4(a1, band[kb], h1);
    }
    // spill H to LDS in C/D layout so the vertical pass can refetch as B
#pragma unroll
    for (int v = 0; v < 8; ++v) {
      const int row = v + rowoff;
      Hb[wave][row][n]      = h0[v];
      Hb[wave][row + 16][n] = h1[v];
    }
    __syncthreads();

    // ---- vertical pass: O(16x16) = Sv(16x28) * H(28x16)
    v8f o = {0.f, 0.f, 0.f, 0.f, 0.f, 0.f, 0.f, 0.f};
#pragma unroll
    for (int kb = 0; kb < 7; ++kb) {
      const int k0 = kb * 4 + khalf;
      const int k1 = k0 + 1;
      v2f b;
      b.x = Hb[wave][k0][n];
      b.y = Hb[wave][k1][n];
      o = wmma4(band[kb], b, o);
    }
    O[m] = o;
    __syncthreads();   // Hb reused by next map
  }

  // ---- per-element SSIM + fused |p-t| over this tile's own 16x16 center
  const float inv121 = 1.0f / 121.0f;
  float ssimAcc = 0.0f, absAcc = 0.0f;
#pragma unroll
  for (int v = 0; v < 8; ++v) {
    const int i = v + rowoff;            // row in tile
    const float mu_x = O[0][v] * inv121;
    const float mu_y = O[1][v] * inv121;
    const float ex2  = O[2][v] * inv121;
    const float ey2  = O[3][v] * inv121;
    const float exy  = O[4][v] * inv121;
    const float sx = ex2 - mu_x * mu_x;
    const float sy = ey2 - mu_y * mu_y;
    const float sxy = exy - mu_x * mu_y;
    const float num = (2.0f * mu_x * mu_y + SSIM_C1) * (2.0f * sxy + SSIM_C2);
    const float den = (mu_x * mu_x + mu_y * mu_y + SSIM_C1) * (sx + sy + SSIM_C2);
    ssimAcc += num / den;
    absAcc += fabsf(Pp[wave][i + 5][n + 5] - Pt[wave][i + 5][n + 5]);
  }

  // ---- block reduction -> global f64 atomics
  red[threadIdx.x] = ssimAcc;
  __syncthreads();
  for (int s = BLOCK_THREADS / 2; s > 0; s >>= 1) {
    if (threadIdx.x < s) red[threadIdx.x] += red[threadIdx.x + s];
    __syncthreads();
  }
  if (threadIdx.x == 0) atomicAdd(&acc[1], (double)red[0]);
  __syncthreads();
  red[threadIdx.x] = absAcc;
  __syncthreads();
  for (int s = BLOCK_THREADS / 2; s > 0; s >>= 1) {
    if (threadIdx.x < s) red[threadIdx.x] += red[threadIdx.x + s];
    __syncthreads();
  }
  if (threadIdx.x == 0) atomicAdd(&acc[0], (double)red[0]);
}

// ---------------------------------------------------------------------------
// Kernel: sampled importance-weighted MSE at N=4096 positions.
// ---------------------------------------------------------------------------
__global__ __launch_bounds__(256) void sampled_kernel(
    const float* __restrict__ pred, const float* __restrict__ targ,
    const int* __restrict__ pos, const float* __restrict__ imp,
    double* __restrict__ acc) {
  __shared__ float red[256];
  const int nidx = blockIdx.x * 256 + threadIdx.x;
  const int u = pos[2 * nidx];
  const int v = pos[2 * nidx + 1];
  const size_t pix = (size_t)u * HW + v;

  float mse = 0.0f;
#pragma unroll 4
  for (int p = 0; p < NPLANES; ++p) {
    const float d = pred[(size_t)p * PLANE + pix] - targ[(size_t)p * PLANE + pix];
    mse += d * d;
  }
  mse *= (1.0f / 48.0f);

  float w = 0.0f;
#pragma unroll 4
  for (int b = 0; b < 16; ++b)
    w += 1.0f / (imp[(size_t)b * PLANE + pix] + 0.1f);
  w *= (1.0f / 16.0f);

  red[threadIdx.x] = w * mse;
  __syncthreads();
  for (int s = 128; s > 0; s >>= 1) {
    if (threadIdx.x < s) red[threadIdx.x] += red[threadIdx.x + s];
    __syncthreads();
  }
  if (threadIdx.x == 0) atomicAdd(&acc[2], (double)red[0]);
}

// ---------------------------------------------------------------------------
// Kernel: combine accumulators into the 5-scalar output tuple.
// ---------------------------------------------------------------------------
__global__ void finalize_kernel(const double* __restrict__ acc,
                                float* __restrict__ out) {
  if (threadIdx.x == 0) {
    const float perceptual = (float)(acc[0] / TOTAL_ELEMS);
    const float structural = (float)(1.0 - acc[1] / TOTAL_ELEMS);
    const float sampled    = (float)(acc[2] / 4096.0);
    const float temporal   = 0.0f;
    out[0] = 0.3f * sampled + 0.4f * perceptual + 0.2f * structural +
             0.1f * temporal;
    out[1] = sampled;
    out[2] = perceptual;
    out[3] = structural;
    out[4] = temporal;
  }
}

extern "C" void kernel_launch(void* const* d_in, const int* in_sizes, int n_in,
                              void* d_out, int out_size, void* d_ws,
                              size_t ws_size, hipStream_t stream) {
  const float* pred = (const float*)d_in[0];
  const float* targ = (const float*)d_in[1];
  const int*   pos  = (const int*)d_in[2];
  const float* imp  = (const float*)d_in[3];
  float* out = (float*)d_out;
  double* acc = (double*)d_ws;

  init_accum<<<1, 32, 0, stream>>>(acc);
  ssim_tile_kernel<<<TOTAL_TILES / WAVES_PER_BLOCK, BLOCK_THREADS, 0, stream>>>(
      pred, targ, acc);
  sampled_kernel<<<4096 / 256, 256, 0, stream>>>(pred, targ, pos, imp, acc);
  finalize_kernel<<<1, 32, 0, stream>>>(acc, out);
}